// Nystroformer_Attention_91336774517470
// MI455X (gfx1250) — compile-verified
//
#include <hip/hip_runtime.h>
#include <hip/hip_bf16.h>

typedef __attribute__((ext_vector_type(2))) float v2f;
typedef __attribute__((ext_vector_type(8))) float v8f;

#define B_    2
#define N_    2048
#define C_    768
#define C3_   2304
#define H_    12
#define D_    64
#define M_    64
#define L_    32
#define BH_   24
#define ROWS_ 4096        // B_*N_

__device__ __forceinline__ v8f wmma4(v2f a, v2f b, v8f c) {
  // D = A(16x4 f32) * B(4x16 f32) + C   -> v_wmma_f32_16x16x4_f32
  return __builtin_amdgcn_wmma_f32_16x16x4_f32(false, a, false, b, (short)0, c, false, false);
}

// ---------------- qkv = x @ w_qkv^T, scattered to head-major q/k/v, q scaled ---
__global__ __launch_bounds__(32) void qkv_kernel(const float* __restrict__ x,
                                                 const float* __restrict__ wqkv,
                                                 float* __restrict__ q,
                                                 float* __restrict__ k,
                                                 float* __restrict__ v) {
  const int lane = threadIdx.x & 31;
  const int half = lane >> 4, l16 = lane & 15;
  const int col0 = blockIdx.x * 16;
  const int row0 = blockIdx.y * 16;
  const float* arow = x    + (size_t)(row0 + l16) * C_ + 2 * half;
  const float* brow = wqkv + (size_t)(col0 + l16) * C_ + 2 * half;
  v8f acc = {};
#pragma unroll 4
  for (int k0 = 0; k0 < C_; k0 += 4) {
    v2f a = *reinterpret_cast<const v2f*>(arow + k0);
    v2f b = *reinterpret_cast<const v2f*>(brow + k0);
    acc = wmma4(a, b, acc);
  }
  const int col = col0 + l16;
  const int t   = col / C_;          // 0=q 1=k 2=v
  const int cc  = col % C_;
  const int h   = cc >> 6, dd = cc & 63;
  float* dst    = (t == 0) ? q : (t == 1) ? k : v;
  const float s = (t == 0) ? 0.125f : 1.0f;   // d^-0.5 = 64^-0.5
#pragma unroll
  for (int r = 0; r < 8; ++r) {
    int row = row0 + r + 8 * half;
    int bb  = row >> 11, n = row & (N_ - 1);
    dst[(((size_t)(bb * H_ + h) * N_ + n) << 6) + dd] = acc[r] * s;
  }
}

// ---------------- landmark means over l=32 groups --------------------------
__global__ void landmark_kernel(const float* __restrict__ q, const float* __restrict__ k,
                                float* __restrict__ ql, float* __restrict__ kl) {
  int idx = blockIdx.x * 256 + threadIdx.x;
  const int total = BH_ * M_ * D_;
  if (idx >= 2 * total) return;
  int which = idx / total;
  int rem   = idx % total;
  int bh = rem >> 12, j = (rem >> 6) & 63, dd = rem & 63;
  const float* src = which ? k : q;
  const float* base = src + ((size_t)bh * N_ + (size_t)j * L_) * D_ + dd;
  float s = 0.0f;
#pragma unroll 8
  for (int i = 0; i < L_; ++i) s += base[(size_t)i * D_];
  (which ? kl : ql)[rem] = s * (1.0f / L_);
}

// ---------------- batched GEMM NT: O[b] = A[b] * B[b]^T --------------------
__global__ __launch_bounds__(32) void gemm_nt(const float* __restrict__ A, int lda, long long As,
                                              const float* __restrict__ Bm, int ldb, long long Bs,
                                              float* __restrict__ O, int ldo, long long Os, int K) {
  const int lane = threadIdx.x & 31;
  const int half = lane >> 4, l16 = lane & 15;
  const int col0 = blockIdx.x * 16, row0 = blockIdx.y * 16, bz = blockIdx.z;
  const float* arow = A  + (size_t)bz * As + (size_t)(row0 + l16) * lda + 2 * half;
  const float* brow = Bm + (size_t)bz * Bs + (size_t)(col0 + l16) * ldb + 2 * half;
  v8f acc = {};
#pragma unroll 4
  for (int k0 = 0; k0 < K; k0 += 4) {
    v2f a = *reinterpret_cast<const v2f*>(arow + k0);
    v2f b = *reinterpret_cast<const v2f*>(brow + k0);
    acc = wmma4(a, b, acc);
  }
  float* orow = O + (size_t)bz * Os;
#pragma unroll
  for (int r = 0; r < 8; ++r)
    orow[(size_t)(row0 + r + 8 * half) * ldo + col0 + l16] = acc[r];
}

// ---------------- batched GEMM NN with epilogue: O = c0*I + c1*(A@B) -------
__global__ __launch_bounds__(32) void gemm_nn(const float* __restrict__ A, int lda, long long As,
                                              const float* __restrict__ Bm, int ldb, long long Bs,
                                              float* __restrict__ O, int ldo, long long Os,
                                              int K, float c0, float c1) {
  const int lane = threadIdx.x & 31;
  const int half = lane >> 4, l16 = lane & 15;
  const int col0 = blockIdx.x * 16, row0 = blockIdx.y * 16, bz = blockIdx.z;
  const float* arow  = A  + (size_t)bz * As + (size_t)(row0 + l16) * lda + 2 * half;
  const float* bbase = Bm + (size_t)bz * Bs + col0 + l16;
  v8f acc = {};
#pragma unroll 4
  for (int k0 = 0; k0 < K; k0 += 4) {
    int kk = k0 + 2 * half;
    v2f a = *reinterpret_cast<const v2f*>(arow + k0);
    v2f b = { bbase[(size_t)kk * ldb], bbase[(size_t)(kk + 1) * ldb] };
    acc = wmma4(a, b, acc);
  }
  float* orow = O + (size_t)bz * Os;
#pragma unroll
  for (int r = 0; r < 8; ++r) {
    int i = row0 + r + 8 * half, j = col0 + l16;
    float vvv = c1 * acc[r] + ((i == j) ? c0 : 0.0f);
    orow[(size_t)i * ldo + j] = vvv;
  }
}

// ---------------- big attn GEMM: O[bh] = attn1[bh] @ invattn3[bh] ----------
// 16x64 strip per wave (4 accumulators share one A fragment); NT streaming
// stores so the 403MB output doesn't evict L2-resident operands.
__global__ __launch_bounds__(32) void attn_gemm(const float* __restrict__ A,   // (BH,N,64)
                                                const float* __restrict__ Bm,  // (BH,64,N)
                                                float* __restrict__ O) {       // (BH,N,N)
  const int lane = threadIdx.x & 31;
  const int half = lane >> 4, l16 = lane & 15;
  const int col0 = blockIdx.x * 64, row0 = blockIdx.y * 16, bz = blockIdx.z;
  const float* arow  = A  + (size_t)bz * ((long long)N_ * M_) + (size_t)(row0 + l16) * M_ + 2 * half;
  const float* bbase = Bm + (size_t)bz * ((long long)M_ * N_) + col0 + l16;
  v8f acc0 = {}, acc1 = {}, acc2 = {}, acc3 = {};
#pragma unroll 4
  for (int k0 = 0; k0 < M_; k0 += 4) {
    int kk = k0 + 2 * half;
    v2f a = *reinterpret_cast<const v2f*>(arow + k0);
    const float* b0 = bbase + (size_t)kk * N_;
    const float* b1 = bbase + (size_t)(kk + 1) * N_;
    v2f bb0 = { b0[0],  b1[0]  };
    v2f bb1 = { b0[16], b1[16] };
    v2f bb2 = { b0[32], b1[32] };
    v2f bb3 = { b0[48], b1[48] };
    acc0 = wmma4(a, bb0, acc0);
    acc1 = wmma4(a, bb1, acc1);
    acc2 = wmma4(a, bb2, acc2);
    acc3 = wmma4(a, bb3, acc3);
  }
  float* orow = O + (size_t)bz * ((long long)N_ * N_);
#pragma unroll
  for (int r = 0; r < 8; ++r) {
    size_t base = (size_t)(row0 + r + 8 * half) * N_ + col0 + l16;
    __builtin_nontemporal_store(acc0[r], &orow[base]);
    __builtin_nontemporal_store(acc1[r], &orow[base + 16]);
    __builtin_nontemporal_store(acc2[r], &orow[base + 32]);
    __builtin_nontemporal_store(acc3[r], &orow[base + 48]);
  }
}

// ---------------- row softmax, in place ------------------------------------
__global__ __launch_bounds__(256) void softmax_rows(float* __restrict__ p, int len) {
  __shared__ float red[256];
  float* row = p + (size_t)blockIdx.x * len;
  const int t = threadIdx.x;
  float mx = -3.0e38f;
  for (int i = t; i < len; i += 256) mx = fmaxf(mx, row[i]);
  red[t] = mx; __syncthreads();
  for (int s = 128; s > 0; s >>= 1) { if (t < s) red[t] = fmaxf(red[t], red[t + s]); __syncthreads(); }
  mx = red[0]; __syncthreads();
  float sum = 0.0f;
  for (int i = t; i < len; i += 256) { float e = __expf(row[i] - mx); row[i] = e; sum += e; }
  red[t] = sum; __syncthreads();
  for (int s = 128; s > 0; s >>= 1) { if (t < s) red[t] += red[t + s]; __syncthreads(); }
  float inv = 1.0f / red[0];
  for (int i = t; i < len; i += 256) row[i] *= inv;
}

// ---------------- pinv helpers ---------------------------------------------
__global__ void init_scal(float* scal) { if (threadIdx.x < 2) scal[threadIdx.x] = 0.0f; }

__global__ __launch_bounds__(64) void pinv_sums(const float* __restrict__ attn2, float* scal) {
  __shared__ float rs[64], cs[64];
  const int bh = blockIdx.x, i = threadIdx.x;
  const float* a = attn2 + (size_t)bh * 4096;
  float r = 0.0f, c = 0.0f;
#pragma unroll 8
  for (int j = 0; j < 64; ++j) { r += fabsf(a[i * 64 + j]); c += fabsf(a[j * 64 + i]); }
  rs[i] = r; cs[i] = c; __syncthreads();
  for (int s = 32; s > 0; s >>= 1) {
    if (i < s) { rs[i] = fmaxf(rs[i], rs[i + s]); cs[i] = fmaxf(cs[i], cs[i + s]); }
    __syncthreads();
  }
  if (i == 0) {  // values are >=0 -> uint ordering == float ordering
    atomicMax(reinterpret_cast<unsigned int*>(&scal[0]), __float_as_uint(rs[0]));
    atomicMax(reinterpret_cast<unsigned int*>(&scal[1]), __float_as_uint(cs[0]));
  }
}

__global__ void pinv_init_z(const float* __restrict__ attn2, const float* __restrict__ scal,
                            float* __restrict__ z) {
  int idx = blockIdx.x * 256 + threadIdx.x;
  if (idx >= BH_ * 4096) return;
  int bh = idx >> 12, i = (idx >> 6) & 63, j = idx & 63;
  float denom = scal[0] * scal[1];
  z[idx] = attn2[(size_t)bh * 4096 + j * 64 + i] / denom;
}

__global__ void eye_minus(const float* __restrict__ src, float* __restrict__ dst, float c0) {
  int idx = blockIdx.x * 256 + threadIdx.x;
  if (idx >= BH_ * 4096) return;
  int i = (idx >> 6) & 63, j = idx & 63;
  dst[idx] = ((i == j) ? c0 : 0.0f) - src[idx];
}

// ---------------- final projection with head gather + bias -----------------
__global__ __launch_bounds__(32) void proj_kernel(const float* __restrict__ outh,
                                                  const float* __restrict__ wproj,
                                                  const float* __restrict__ bproj,
                                                  float* __restrict__ out) {
  const int lane = threadIdx.x & 31;
  const int half = lane >> 4, l16 = lane & 15;
  const int col0 = blockIdx.x * 16, row0 = blockIdx.y * 16;
  const int row = row0 + l16;
  const int bb = row >> 11, n = row & (N_ - 1);
  const float* brow = wproj + (size_t)(col0 + l16) * C_ + 2 * half;
  v8f acc = {};
#pragma unroll 4
  for (int k0 = 0; k0 < C_; k0 += 4) {
    int kk = k0 + 2 * half;          // even, so kk and kk+1 share a head block
    int h = kk >> 6, dd = kk & 63;
    v2f a = *reinterpret_cast<const v2f*>(outh + (((size_t)(bb * H_ + h) * N_ + n) << 6) + dd);
    v2f b = *reinterpret_cast<const v2f*>(brow + k0);
    acc = wmma4(a, b, acc);
  }
  const float bias = bproj[col0 + l16];
#pragma unroll
  for (int r = 0; r < 8; ++r)
    __builtin_nontemporal_store(acc[r] + bias, &out[(size_t)(row0 + r + 8 * half) * C_ + col0 + l16]);
}

extern "C" void kernel_launch(void* const* d_in, const int* in_sizes, int n_in,
                              void* d_out, int out_size, void* d_ws, size_t ws_size,
                              hipStream_t stream) {
  (void)in_sizes; (void)n_in; (void)out_size; (void)ws_size;
  const float* x     = (const float*)d_in[0];
  const float* wqkv  = (const float*)d_in[1];
  const float* wproj = (const float*)d_in[2];
  const float* bproj = (const float*)d_in[3];
  float* out = (float*)d_out;                       // [0, 3145728): out; then attn
  float* ws  = (float*)d_ws;

  const long long BIG = 3145728;   // BH_*N_*D_ = 24*2048*64
  const long long SML = 98304;     // BH_*64*64
  float* q        = ws;
  float* k        = q + BIG;
  float* v        = k + BIG;
  float* attn1    = v + BIG;          // sim1 -> softmax in place (BH, N, 64)
  float* attn3    = attn1 + BIG;      // sim3 (BH, 64, N)
  float* outh     = attn3 + BIG;      // (BH, N, 64)
  float* invattn3 = outh + BIG;       // z @ attn3 (BH, 64, N)
  float* ql       = invattn3 + BIG;
  float* kl       = ql + SML;
  float* attn2    = kl + SML;
  float* zA       = attn2 + SML;
  float* zB       = zA + SML;
  float* xz       = zB + SML;
  float* wA       = xz + SML;
  float* wB       = wA + SML;
  float* a3v      = wB + SML;         // attn3 @ v (BH,64,64)
  float* i3v      = a3v + SML;        // z @ a3v  (BH,64,64)
  float* scal     = i3v + SML;        // 2 floats: max row/col sums

  const dim3 blk(32);
  const long long Sqn = (long long)N_ * D_;   // q/k/v per-head stride
  const long long Slm = (long long)M_ * D_;   // landmark per-head stride
  const long long Sm  = (long long)M_ * M_;   // 64x64 stride
  const long long Snm = (long long)N_ * M_;   // attn1 per-head stride
  const long long Smn = (long long)M_ * N_;   // attn3 per-head stride

  // 1) qkv projection, scatter to head-major, scale q
  qkv_kernel<<<dim3(C3_ / 16, ROWS_ / 16), blk, 0, stream>>>(x, wqkv, q, k, v);
  // 2) landmarks
  landmark_kernel<<<dim3((2 * BH_ * M_ * D_ + 255) / 256), dim3(256), 0, stream>>>(q, k, ql, kl);
  // 3) sims (all X @ Y^T)
  gemm_nt<<<dim3(M_ / 16, N_ / 16, BH_), blk, 0, stream>>>(q,  D_, Sqn, kl, D_, Slm, attn1, M_, Snm, D_);
  gemm_nt<<<dim3(M_ / 16, M_ / 16, BH_), blk, 0, stream>>>(ql, D_, Slm, kl, D_, Slm, attn2, M_, Sm,  D_);
  gemm_nt<<<dim3(N_ / 16, M_ / 16, BH_), blk, 0, stream>>>(ql, D_, Slm, k,  D_, Sqn, attn3, N_, Smn, D_);
  // 4) softmaxes
  softmax_rows<<<dim3(BH_ * N_), dim3(256), 0, stream>>>(attn1, M_);
  softmax_rows<<<dim3(BH_ * M_), dim3(256), 0, stream>>>(attn2, M_);
  softmax_rows<<<dim3(BH_ * M_), dim3(256), 0, stream>>>(attn3, N_);
  // 5) Moore-Penrose pinv of attn2
  init_scal<<<1, 32, 0, stream>>>(scal);
  pinv_sums<<<dim3(BH_), dim3(64), 0, stream>>>(attn2, scal);
  pinv_init_z<<<dim3(BH_ * 4096 / 256), dim3(256), 0, stream>>>(attn2, scal, zA);
  float* z = zA; float* z2 = zB;
  for (int it = 0; it < 6; ++it) {
    gemm_nn<<<dim3(4, 4, BH_), blk, 0, stream>>>(attn2, M_, Sm, z,  M_, Sm, xz, M_, Sm, M_, 0.0f,  1.0f);
    eye_minus<<<dim3(BH_ * 4096 / 256), dim3(256), 0, stream>>>(xz, wA, 7.0f);
    gemm_nn<<<dim3(4, 4, BH_), blk, 0, stream>>>(xz, M_, Sm, wA, M_, Sm, wB, M_, Sm, M_, 15.0f, -1.0f);
    gemm_nn<<<dim3(4, 4, BH_), blk, 0, stream>>>(xz, M_, Sm, wB, M_, Sm, wA, M_, Sm, M_, 13.0f, -1.0f);
    gemm_nn<<<dim3(4, 4, BH_), blk, 0, stream>>>(z,  M_, Sm, wA, M_, Sm, z2, M_, Sm, M_, 0.0f,  0.25f);
    float* tmp = z; z = z2; z2 = tmp;
  }
  // 6) a3v = attn3 @ v ; i3v = z @ a3v ; outh = attn1 @ i3v
  gemm_nn<<<dim3(4, 4, BH_), blk, 0, stream>>>(attn3, N_, Smn, v,   D_, Sqn, a3v,  D_, Slm, N_, 0.0f, 1.0f);
  gemm_nn<<<dim3(4, 4, BH_), blk, 0, stream>>>(z,     M_, Sm,  a3v, D_, Slm, i3v,  D_, Slm, M_, 0.0f, 1.0f);
  gemm_nn<<<dim3(4, N_ / 16, BH_), blk, 0, stream>>>(attn1, M_, Snm, i3v, D_, Slm, outh, D_, Sqn, M_, 0.0f, 1.0f);
  // 7) attn = attn1 @ (z @ attn3), streamed into d_out with NT stores
  gemm_nn<<<dim3(N_ / 16, 4, BH_), blk, 0, stream>>>(z, M_, Sm, attn3, N_, Smn, invattn3, N_, Smn, M_, 0.0f, 1.0f);
  attn_gemm<<<dim3(N_ / 64, N_ / 16, BH_), blk, 0, stream>>>(attn1, invattn3, out + (size_t)ROWS_ * C_);
  // 8) final projection + bias
  proj_kernel<<<dim3(C_ / 16, ROWS_ / 16), blk, 0, stream>>>(outh, wproj, bproj, out);
}